// HyperEncoder_12970801234150
// MI455X (gfx1250) — compile-verified
//
#include <hip/hip_runtime.h>

typedef float v2f __attribute__((ext_vector_type(2)));
typedef float v8f __attribute__((ext_vector_type(8)));

#define FDIM 128

// ---------------------------------------------------------------------------
// GEMM: C[M,128] = A[M,128] @ B[128,128], fp32, WMMA f32 16x16x4.
// Block = 256 threads = 8 waves. Each block computes one 16-row strip of C;
// each wave owns one 16x16 tile (n0 = wave*16). Weights + A-strip staged in LDS.
// M must be a multiple of 16 (N=50000, E=10000 both are).
// ---------------------------------------------------------------------------
__global__ __launch_bounds__(256)
void gemm_wmma_f32(const float* __restrict__ A, const float* __restrict__ B,
                   float* __restrict__ C, int M) {
    __shared__ float sB[FDIM * FDIM];   // 64 KB weights
    __shared__ float sA[16 * FDIM];     // 8 KB A strip

    const int t  = threadIdx.x;
    const int m0 = blockIdx.x * 16;

    // Load weights: 16384 floats = 4096 float4; 16 per thread.
    const float4* B4  = (const float4*)B;
    float4*       sB4 = (float4*)sB;
#pragma unroll
    for (int i = 0; i < 16; ++i) sB4[t + 256 * i] = B4[t + 256 * i];

    // Load A strip: 2048 floats = 512 float4; 2 per thread.
    const float4* A4  = (const float4*)(A + (size_t)m0 * FDIM);
    float4*       sA4 = (float4*)sA;
    sA4[t]       = A4[t];
    sA4[t + 256] = A4[t + 256];

    __syncthreads();

    const int wave = t >> 5;
    const int lane = t & 31;
    const int n0   = wave * 16;
    const int l    = lane & 15;        // row-in-tile / col-in-tile
    const int hi   = lane >> 4;        // 0: K pair {0,1}; 1: K pair {2,3}
    const int koff = hi * 2;

    v8f acc = {};
#pragma unroll
    for (int k0 = 0; k0 < FDIM; k0 += 4) {
        v2f a, b;
        // A 16x4 layout: lanes 0-15 hold K=k0+0/k0+1, lanes 16-31 K=k0+2/k0+3
        a.x = sA[l * FDIM + k0 + koff];
        a.y = sA[l * FDIM + k0 + koff + 1];
        // B 4x16 layout mirrors A striping across the half-waves
        b.x = sB[(k0 + koff) * FDIM + n0 + l];
        b.y = sB[(k0 + koff + 1) * FDIM + n0 + l];
        acc = __builtin_amdgcn_wmma_f32_16x16x4_f32(
            /*neg_a=*/false, a, /*neg_b=*/false, b,
            /*c_mod=*/(short)0, acc, /*reuse_a=*/false, /*reuse_b=*/false);
    }

    // C/D layout: VGPR r -> row m0+r (lanes 0-15) / m0+8+r (lanes 16-31), col n0+lane%16
    const int rbase = m0 + hi * 8;
    const int col   = n0 + l;
#pragma unroll
    for (int r = 0; r < 8; ++r)
        C[(size_t)(rbase + r) * FDIM + col] = acc[r];
}

// ---------------------------------------------------------------------------
// Segment scatter-add: for each pair p: acc[didx[p]] += src[sidx[p]] (128-wide),
// cnt[didx[p]] += 1. Block = 128 threads (one per feature), several pairs/block.
// ---------------------------------------------------------------------------
__global__ __launch_bounds__(128)
void scatter_add_rows(const float* __restrict__ src, const int* __restrict__ sidx,
                      const int* __restrict__ didx, float* __restrict__ acc,
                      float* __restrict__ cnt, int npairs, int pairs_per_block) {
    const int t  = threadIdx.x;
    int p        = blockIdx.x * pairs_per_block;
    int pend     = p + pairs_per_block;
    if (pend > npairs) pend = npairs;
    for (; p < pend; ++p) {
        const int s = sidx[p];
        const int d = didx[p];
        atomicAdd(&acc[(size_t)d * FDIM + t], src[(size_t)s * FDIM + t]);
        if (t == 0) atomicAdd(&cnt[d], 1.0f);
    }
}

// ---------------------------------------------------------------------------
// mean + optional pre-activation output + PReLU.  Grid = rows, block = 128.
// ---------------------------------------------------------------------------
__global__ __launch_bounds__(128)
void finalize_mean_prelu(const float* __restrict__ acc, const float* __restrict__ cnt,
                         const float* __restrict__ a_ptr, float* __restrict__ out_act,
                         float* __restrict__ out_pre) {
    const int r = blockIdx.x;
    const int t = threadIdx.x;
    const float c = fmaxf(cnt[r], 1.0f);
    const float v = acc[(size_t)r * FDIM + t] / c;
    if (out_pre) out_pre[(size_t)r * FDIM + t] = v;
    const float a = *a_ptr;
    out_act[(size_t)r * FDIM + t] = (v >= 0.0f) ? v : a * v;
}

// c = mean_e + mean_n
__global__ __launch_bounds__(128)
void finalize_components(const float* __restrict__ accE, const float* __restrict__ cntE,
                         const float* __restrict__ accN, const float* __restrict__ cntN,
                         float* __restrict__ out) {
    const int r = blockIdx.x;
    const int t = threadIdx.x;
    const float ce = fmaxf(cntE[r], 1.0f);
    const float cn = fmaxf(cntN[r], 1.0f);
    out[(size_t)r * FDIM + t] =
        accE[(size_t)r * FDIM + t] / ce + accN[(size_t)r * FDIM + t] / cn;
}

// ---------------------------------------------------------------------------
extern "C" void kernel_launch(void* const* d_in, const int* in_sizes, int n_in,
                              void* d_out, int out_size, void* d_ws, size_t ws_size,
                              hipStream_t stream) {
    (void)n_in; (void)ws_size;

    const float* x     = (const float*)d_in[0];
    const int*   hei   = (const int*)d_in[1];   // [2, NNZ]: row0 = node idx, row1 = edge idx
    const int*   heci  = (const int*)d_in[2];   // [2, E]:   row0 = edge idx, row1 = comp idx
    const int*   nci   = (const int*)d_in[3];   // [2, N]:   row0 = node idx, row1 = comp idx
    const float* Wne0  = (const float*)d_in[7];
    const float* Wen0  = (const float*)d_in[8];
    const float* Wec1  = (const float*)d_in[13];
    const float* Wnc1  = (const float*)d_in[14];
    const float* Wne1  = (const float*)d_in[11];
    const float* Wen1  = (const float*)d_in[12];
    const float* prelu = (const float*)d_in[15];

    const int N   = in_sizes[0] / FDIM;
    const int NNZ = in_sizes[1] / 2;
    const int E   = in_sizes[2] / 2;
    const int C   = out_size / FDIM - N - E;

    const int* ni   = hei;            // node endpoint per pair
    const int* ei   = hei + NNZ;      // edge endpoint per pair
    const int* ec_e = heci;
    const int* ec_c = heci + E;
    const int* nc_n = nci;
    const int* nc_c = nci + N;

    float* out_x = (float*)d_out;                       // [N,128] (also holds layer-1 x)
    float* out_e = out_x + (size_t)N * FDIM;            // [E,128]
    float* out_c = out_e + (size_t)E * FDIM;            // [C,128]

    // workspace layout (floats)
    float* bufA  = (float*)d_ws;                        // N*128 GEMM output (node-sized)
    float* Nacc  = bufA  + (size_t)N * FDIM;            // N*128 node accumulator
    float* Eacc  = Nacc  + (size_t)N * FDIM;            // E*128 edge accumulator
    float* Eact  = Eacc  + (size_t)E * FDIM;            // E*128 prelu(e)
    float* Eg    = Eact  + (size_t)E * FDIM;            // E*128 GEMM output (edge-sized)
    float* CaccE = Eg    + (size_t)E * FDIM;            // C*128
    float* CaccN = CaccE + (size_t)C * FDIM;            // C*128
    float* cntE  = CaccN + (size_t)C * FDIM;            // E
    float* cntN  = cntE + E;                            // N
    float* cntCE = cntN + N;                            // C
    float* cntCN = cntCE + C;                           // C

    const int PPB = 4;  // pairs per scatter block
    dim3 b128(128), b256(256);
    dim3 gN(N / 16), gE(E / 16);
    dim3 gScatNNZ((NNZ + PPB - 1) / PPB), gScatE((E + PPB - 1) / PPB),
         gScatN((N + PPB - 1) / PPB);

    // ======================= layer 0 =======================
    hipMemsetAsync(Eacc, 0, (size_t)E * FDIM * sizeof(float), stream);
    hipMemsetAsync(cntE, 0, (size_t)E * sizeof(float), stream);
    gemm_wmma_f32<<<gN, b256, 0, stream>>>(x, Wne0, bufA, N);
    scatter_add_rows<<<gScatNNZ, b128, 0, stream>>>(bufA, ni, ei, Eacc, cntE, NNZ, PPB);
    finalize_mean_prelu<<<dim3(E), b128, 0, stream>>>(Eacc, cntE, prelu, Eact, nullptr);

    hipMemsetAsync(Nacc, 0, (size_t)N * FDIM * sizeof(float), stream);
    hipMemsetAsync(cntN, 0, (size_t)N * sizeof(float), stream);
    gemm_wmma_f32<<<gE, b256, 0, stream>>>(Eact, Wen0, Eg, E);
    scatter_add_rows<<<gScatNNZ, b128, 0, stream>>>(Eg, ei, ni, Nacc, cntN, NNZ, PPB);
    // x1 = prelu(n) -> stored in out_x (layer-1 components are dead in the reference)
    finalize_mean_prelu<<<dim3(N), b128, 0, stream>>>(Nacc, cntN, prelu, out_x, nullptr);

    // ======================= layer 1 =======================
    hipMemsetAsync(Eacc, 0, (size_t)E * FDIM * sizeof(float), stream);
    hipMemsetAsync(cntE, 0, (size_t)E * sizeof(float), stream);
    gemm_wmma_f32<<<gN, b256, 0, stream>>>(out_x, Wne1, bufA, N);
    scatter_add_rows<<<gScatNNZ, b128, 0, stream>>>(bufA, ni, ei, Eacc, cntE, NNZ, PPB);
    // e_out (pre-activation) -> d_out; Eact = prelu(e)
    finalize_mean_prelu<<<dim3(E), b128, 0, stream>>>(Eacc, cntE, prelu, Eact, out_e);

    hipMemsetAsync(Nacc, 0, (size_t)N * FDIM * sizeof(float), stream);
    hipMemsetAsync(cntN, 0, (size_t)N * sizeof(float), stream);
    hipMemsetAsync(CaccE, 0, (size_t)C * FDIM * sizeof(float), stream);
    hipMemsetAsync(CaccN, 0, (size_t)C * FDIM * sizeof(float), stream);
    hipMemsetAsync(cntCE, 0, (size_t)C * sizeof(float), stream);
    hipMemsetAsync(cntCN, 0, (size_t)C * sizeof(float), stream);

    gemm_wmma_f32<<<gE, b256, 0, stream>>>(Eact, Wen1, Eg, E);
    scatter_add_rows<<<gScatNNZ, b128, 0, stream>>>(Eg, ei, ni, Nacc, cntN, NNZ, PPB);

    gemm_wmma_f32<<<gE, b256, 0, stream>>>(Eact, Wec1, Eg, E);      // reuse Eg
    scatter_add_rows<<<gScatE, b128, 0, stream>>>(Eg, ec_e, ec_c, CaccE, cntCE, E, PPB);

    gemm_wmma_f32<<<gN, b256, 0, stream>>>(out_x, Wnc1, bufA, N);   // reads layer-1 x
    scatter_add_rows<<<gScatN, b128, 0, stream>>>(bufA, nc_n, nc_c, CaccN, cntCN, N, PPB);

    finalize_components<<<dim3(C), b128, 0, stream>>>(CaccE, cntCE, CaccN, cntCN, out_c);
    // final x = prelu(n) — overwrites layer-1 x, must be the last user of out_x
    finalize_mean_prelu<<<dim3(N), b128, 0, stream>>>(Nacc, cntN, prelu, out_x, nullptr);
}